// decoderTriNet_75419625717943
// MI455X (gfx1250) — compile-verified
//
#include <hip/hip_runtime.h>

// decoderTriNet on MI455X (gfx1250).
//
// Algebraic collapse: relu(tau + b_tri) -> 2nd difference -> dot(c_tri) is a
// triangular-basis evaluation == 2-tap linear interpolation of c_tri at
// u = (tau - t_phi0)*RES, scaled by T_INT.  Per output we need exactly two
// gathered floats, so the kernel is pure data movement (~130 KB total, ~6 ns
// of HBM time at 23.3 TB/s) and launch-latency bound.  WMMA would require a
// 268 MB dense A operand for a 130 KB problem -> structurally wrong here.
//
// CDNA5 path: ASYNCcnt-tracked global->LDS DMA (global_load_async_to_lds_b128)
// stages the 32 KB c_tri table into LDS once per workgroup.  The async op's
// 24-bit IOFFSET is applied to BOTH the LDS address and the global address
// (CDNA5 ISA), and our LDS/global offsets advance in
// lockstep, so all 8 copies share one LDS-addr VGPR and one global-addr VGPR
// pair, varying only the immediate offset — no per-iteration address VALU.
// Each wave drains its own ASYNCcnt, then the workgroup barrier publishes all
// waves' LDS writes; the 2-tap gathers then hit LDS banks instead of VMEM.

#define B_DIM   16
#define N_DIM   128
#define K_DIM   8
#define KNOTS   8192                 // N*RES
#define RES_F   64.0f
#define T_INT_F (1.0f / 64.0f)
#define PERIOD  128.0f

// One async 16B copy; `off` immediate applies to both LDS and global sides.
#define ASYNC_CP(lds_v, gaddr_v, off)                                   \
    asm volatile("global_load_async_to_lds_b128 %0, %1, off offset:" #off \
                 :                                                      \
                 : "v"(lds_v), "v"(gaddr_v)                             \
                 : "memory")

__global__ __launch_bounds__(256)
void decoderTriNet_kernel(const float* __restrict__ t_k_hat,  // [B,K]
                          const float* __restrict__ c_tri,    // [KNOTS]
                          const float* __restrict__ b_tri,    // [KNOTS+2]
                          float* __restrict__ out)            // [B,N,K]
{
    __shared__ float lds_c[KNOTS];                 // 32 KB of the 320 KB WGP LDS

    const int tid = threadIdx.x;
    const int b   = blockIdx.x;

    // ---- Stage c_tri into LDS via CDNA5 async global->LDS copies ----------
    // Low 32 bits of a generic pointer to LDS are the LDS byte offset
    // (generic-to-LDS aperture rule).
    const uint32_t lds_v   = (uint32_t)(uintptr_t)&lds_c[0] + (uint32_t)(tid * 16);
    const float*   gaddr_v = c_tri + tid * 4;      // coalesced 4 KB per issue

    ASYNC_CP(lds_v, gaddr_v, 0);
    ASYNC_CP(lds_v, gaddr_v, 4096);
    ASYNC_CP(lds_v, gaddr_v, 8192);
    ASYNC_CP(lds_v, gaddr_v, 12288);
    ASYNC_CP(lds_v, gaddr_v, 16384);
    ASYNC_CP(lds_v, gaddr_v, 20480);
    ASYNC_CP(lds_v, gaddr_v, 24576);
    ASYNC_CP(lds_v, gaddr_v, 28672);

    asm volatile("s_wait_asynccnt 0" ::: "memory");
    __syncthreads();

    // t_phi[0] = -b_tri[1]  (uniform address -> scalar load, L2-resident)
    const float t0 = -b_tri[1];

    // ---- 4 outputs per thread: o = n*K + k, contiguous -> b128 store ------
    float res[4];
#pragma unroll
    for (int q = 0; q < 4; ++q) {
        const int   o  = tid * 4 + q;
        const int   n  = o >> 3;
        const int   k  = o & (K_DIM - 1);
        const float t  = t_k_hat[b * K_DIM + k];

        // remainder(t - n, 128): |t - n| < 128, so one conditional add suffices
        float tau = t - (float)n;
        tau += (tau < 0.0f) ? PERIOD : 0.0f;

        const float u  = (tau - t0) * RES_F;
        const float fu = floorf(u);
        const int   i0 = (int)fu;
        const float fr = u - fu;

        // Boundary hats fall off the knot table -> contribute 0 (exact match
        // to the reference's complete-hat construction with guard knots).
        const float w0 = ((unsigned)i0       < (unsigned)KNOTS) ? lds_c[i0]     : 0.0f;
        const float w1 = ((unsigned)(i0 + 1) < (unsigned)KNOTS) ? lds_c[i0 + 1] : 0.0f;

        res[q] = (w0 * (1.0f - fr) + w1 * fr) * T_INT_F;
    }

    float4* o4 = reinterpret_cast<float4*>(out + (size_t)b * N_DIM * K_DIM);
    o4[tid] = make_float4(res[0], res[1], res[2], res[3]);
}

extern "C" void kernel_launch(void* const* d_in, const int* in_sizes, int n_in,
                              void* d_out, int out_size, void* d_ws, size_t ws_size,
                              hipStream_t stream)
{
    (void)in_sizes; (void)n_in; (void)out_size; (void)d_ws; (void)ws_size;

    const float* t_k_hat = (const float*)d_in[0];   // [16,8]
    const float* c_tri   = (const float*)d_in[1];   // [8192]
    const float* b_tri   = (const float*)d_in[2];   // [8194]
    float*       out     = (float*)d_out;           // [16,128,8]

    decoderTriNet_kernel<<<dim3(B_DIM), dim3(256), 0, stream>>>(
        t_k_hat, c_tri, b_tri, out);
}